// ClusterPool_51342039056603
// MI455X (gfx1250) — compile-verified
//
#include <hip/hip_runtime.h>
#include <math.h>

typedef __attribute__((ext_vector_type(2))) float v2f;
typedef __attribute__((ext_vector_type(8))) float v8f;

#define DEVINL __device__ __forceinline__

// ---------------- problem dims ----------------
constexpr int B0   = 16;
constexpr int CIN  = 64;
constexpr int H    = 96;
constexpr int W    = 96;
constexpr int COUT = 128;
constexpr int BF   = B0 * 4;        // 64 folded images (b, f1, f2)
constexpr int FW   = 48;            // folded spatial (both dims)
constexpr int NPTS = FW * FW;       // 2304 points per folded image
constexpr int CW   = 24;            // pooled spatial
constexpr int M    = CW * CW;       // 576 centers per folded image
constexpr int OH   = 48, OW = 48;   // output spatial
constexpr int NT   = NPTS / 16;     // 144 point tiles
constexpr int MT   = M / 16;        // 36 center tiles
constexpr int OT   = COUT / 16;     // 8 out-channel tiles

// ---------------- workspace layout (in floats) ----------------
constexpr size_t SZ_FEAT = (size_t)BF * NPTS * COUT;   // 18,874,368
constexpr size_t SZ_CEN  = (size_t)BF * M * COUT;      //  4,718,592
constexpr size_t OFF_FEAT = 0;                         // featP; reused as outPre later
constexpr size_t OFF_VAL  = OFF_FEAT + SZ_FEAT;        // valueP
constexpr size_t OFF_CEN  = OFF_VAL  + SZ_FEAT;        // normalized centers
constexpr size_t OFF_VCEN = OFF_CEN  + SZ_CEN;         // value centers
constexpr size_t OFF_POOL = OFF_VCEN + SZ_CEN;         // pooled accumulator
constexpr size_t OFF_CNT  = OFF_POOL + SZ_CEN;         // BF*M counts
constexpr size_t OFF_GN   = OFF_CNT  + (size_t)BF * M; // 32 floats (sum,sumsq per b)
constexpr size_t OFF_IDX  = OFF_GN   + 32;             // BF*NPTS ints

// ---------------- WMMA helper (f32 16x16x4) ----------------
// A: 16x4 f32, lane holds A[m=lane&15][koff..koff+1], koff = (lane<16?0:2)
// B: 4x16 f32, lane holds B[koff..koff+1][n=lane&15]
// D: 16x16 f32, d[v] = D[v + (lane<16?0:8)][lane&15]
DEVINL v8f wmma_f32(v2f a, v2f b, v8f c) {
  return __builtin_amdgcn_wmma_f32_16x16x4_f32(
      /*neg_a=*/false, a, /*neg_b=*/false, b,
      /*c_mod=*/(short)0, c, /*reuse_a=*/false, /*reuse_b=*/false);
}

DEVINL float waveReduceSum(float v) {
#pragma unroll
  for (int off = 16; off >= 1; off >>= 1) v += __shfl_xor(v, off, 32);
  return v;
}

// ---------------- K1: 1x1 convs (feat & value) via WMMA ----------------
__global__ __launch_bounds__(256) void k_gemm_fv(
    const float* __restrict__ x,
    const float* __restrict__ wf, const float* __restrict__ bfp,
    const float* __restrict__ wv, const float* __restrict__ bvp,
    float* __restrict__ featP, float* __restrict__ valP) {
  const int lane = threadIdx.x & 31;
  const int wave = (int)((blockIdx.x * blockDim.x + threadIdx.x) >> 5);
  const int otile = wave & (OT - 1);
  const int nt_bf = wave >> 3;
  const int ntile = nt_bf % NT;
  const int bf    = nt_bf / NT;
  const int b  = bf >> 2, f1 = (bf >> 1) & 1, f2 = bf & 1;
  const int l15 = lane & 15;
  const int hlf = (lane >> 4) & 1;

  // B operand: this lane's folded pixel
  const int n  = ntile * 16 + l15;
  const int pi = n / FW, pj = n % FW;
  const int ph = f1 * FW + pi, pw = f2 * FW + pj;
  const float* xpix = x + (size_t)b * CIN * H * W + (size_t)ph * W + pw;

  // A operands: this lane's output-channel row of the 1x1 weights
  const int oa = otile * 16 + l15;
  const float* wfrow = wf + (size_t)oa * CIN;
  const float* wvrow = wv + (size_t)oa * CIN;

  v8f accf = {};
  v8f accv = {};
  const int koffBase = 2 * hlf;
  for (int k = 0; k < CIN; k += 4) {
    const int ko = k + koffBase;
    v2f af = *(const v2f*)(wfrow + ko);
    v2f av = *(const v2f*)(wvrow + ko);
    v2f bb = { xpix[(size_t)ko * (H * W)], xpix[(size_t)(ko + 1) * (H * W)] };
    accf = wmma_f32(af, bb, accf);
    accv = wmma_f32(av, bb, accv);
  }

  // epilogue: contiguous 8-channel chunk per lane in [bf][n][c] layout
  const int obase = otile * 16 + hlf * 8;
  float* frow = featP + (size_t)(bf * NPTS + n) * COUT + obase;
  float* vrow = valP  + (size_t)(bf * NPTS + n) * COUT + obase;
#pragma unroll
  for (int v = 0; v < 8; ++v) {
    frow[v] = accf[v] + bfp[obase + v];
    vrow[v] = accv[v] + bvp[obase + v];
  }
}

// ---------------- K2: avg-pool centers + L2-normalize centers ----------------
__global__ __launch_bounds__(256) void k_centers(
    const float* __restrict__ featP, const float* __restrict__ valP,
    float* __restrict__ cenN, float* __restrict__ vcen) {
  const int lane = threadIdx.x & 31;
  const int wave = (int)((blockIdx.x * blockDim.x + threadIdx.x) >> 5);
  const int m  = wave % M;
  const int bf = wave / M;
  const int ci = m / CW, cj = m % CW;
  const int c4 = lane * 4;

  const size_t rowBase = (size_t)bf * NPTS * COUT;
  const int n00 = (2 * ci) * FW + 2 * cj;
  const float4 f00 = *(const float4*)(featP + rowBase + (size_t)n00 * COUT + c4);
  const float4 f01 = *(const float4*)(featP + rowBase + (size_t)(n00 + 1) * COUT + c4);
  const float4 f10 = *(const float4*)(featP + rowBase + (size_t)(n00 + FW) * COUT + c4);
  const float4 f11 = *(const float4*)(featP + rowBase + (size_t)(n00 + FW + 1) * COUT + c4);
  float4 cen;
  cen.x = 0.25f * (f00.x + f01.x + f10.x + f11.x);
  cen.y = 0.25f * (f00.y + f01.y + f10.y + f11.y);
  cen.z = 0.25f * (f00.z + f01.z + f10.z + f11.z);
  cen.w = 0.25f * (f00.w + f01.w + f10.w + f11.w);

  const float4 v00 = *(const float4*)(valP + rowBase + (size_t)n00 * COUT + c4);
  const float4 v01 = *(const float4*)(valP + rowBase + (size_t)(n00 + 1) * COUT + c4);
  const float4 v10 = *(const float4*)(valP + rowBase + (size_t)(n00 + FW) * COUT + c4);
  const float4 v11 = *(const float4*)(valP + rowBase + (size_t)(n00 + FW + 1) * COUT + c4);
  float4 vc;
  vc.x = 0.25f * (v00.x + v01.x + v10.x + v11.x);
  vc.y = 0.25f * (v00.y + v01.y + v10.y + v11.y);
  vc.z = 0.25f * (v00.z + v01.z + v10.z + v11.z);
  vc.w = 0.25f * (v00.w + v01.w + v10.w + v11.w);

  float ss = cen.x * cen.x + cen.y * cen.y + cen.z * cen.z + cen.w * cen.w;
  ss = waveReduceSum(ss);                 // full 128-channel sum of squares
  const float scale = 1.0f / fmaxf(sqrtf(ss), 1e-12f);
  cen.x *= scale; cen.y *= scale; cen.z *= scale; cen.w *= scale;

  *(float4*)(cenN + (size_t)(bf * M + m) * COUT + c4) = cen;
  *(float4*)(vcen + (size_t)(bf * M + m) * COUT + c4) = vc;
}

// ---------------- K3: cosine-sim GEMM + argmax via WMMA ----------------
// Per-point normalization is a positive per-column scalar and cannot change
// argmax over centers, so pts = raw feat; only centers are normalized.
// The lane's K-half of its point row (64 floats) is preloaded into 32 v2f
// registers and reused across all 36 center tiles (removes 35/36 of the
// B-operand VMEM traffic in the hot loop).
__global__ __launch_bounds__(256) void k_sim(
    const float* __restrict__ featP, const float* __restrict__ cenN,
    int* __restrict__ idx) {
  const int lane = threadIdx.x & 31;
  const int wave = (int)((blockIdx.x * blockDim.x + threadIdx.x) >> 5);
  const int ntile = wave % NT;
  const int bf    = wave / NT;
  const int l15 = lane & 15;
  const int hlf = (lane >> 4) & 1;
  const int koffBase = 2 * hlf;

  const int n = ntile * 16 + l15;
  const float* prow  = featP + (size_t)(bf * NPTS + n) * COUT + koffBase;
  const float* cbase = cenN  + (size_t)bf * M * COUT + koffBase;

  v2f p[COUT / 4];
#pragma unroll
  for (int kk = 0; kk < COUT / 4; ++kk)
    p[kk] = *(const v2f*)(prow + 4 * kk);

  float best = -3.4e38f;
  int bestm = 0;
  for (int mt = 0; mt < MT; ++mt) {
    const float* crow = cbase + (size_t)(mt * 16 + l15) * COUT;
    v8f acc = {};
#pragma unroll
    for (int kk = 0; kk < COUT / 4; ++kk) {
      v2f a = *(const v2f*)(crow + 4 * kk);
      acc = wmma_f32(a, p[kk], acc);
    }
    // local argmax over this lane's 8 rows (m = mt*16 + hlf*8 + v)
    float lv = acc[0];
    int   lm = mt * 16 + hlf * 8;
#pragma unroll
    for (int v = 1; v < 8; ++v) {
      const float t = acc[v];
      if (t > lv) { lv = t; lm = mt * 16 + hlf * 8 + v; }
    }
    // combine the two row halves held by lane and lane^16 (same point n)
    const float pv = __shfl_xor(lv, 16, 32);
    const int   pm = __shfl_xor(lm, 16, 32);
    if (pv > lv || (pv == lv && pm < lm)) { lv = pv; lm = pm; }
    if (lv > best || (lv == best && lm < bestm)) { best = lv; bestm = lm; }
  }
  if (lane < 16) idx[bf * NPTS + n] = bestm;
}

// ---------------- K4: init pooled=vcen, counts=1, gnbuf=0 ----------------
__global__ __launch_bounds__(256) void k_init(
    const float* __restrict__ vcen, float* __restrict__ pooled,
    float* __restrict__ counts, float* __restrict__ gnbuf) {
  const size_t i = (size_t)blockIdx.x * blockDim.x + threadIdx.x;
  if (i < SZ_CEN) {
    pooled[i] = vcen[i];
  } else if (i < SZ_CEN + (size_t)BF * M) {
    counts[i - SZ_CEN] = 1.0f;
  } else if (i < SZ_CEN + (size_t)BF * M + 32) {
    gnbuf[i - SZ_CEN - (size_t)BF * M] = 0.0f;
  }
}

// ---------------- K5: segment-sum scatter of values ----------------
__global__ __launch_bounds__(256) void k_scatter(
    const float* __restrict__ valP, const int* __restrict__ idx,
    float* __restrict__ pooled, float* __restrict__ counts) {
  const int lane = threadIdx.x & 31;
  const int wave = (int)((blockIdx.x * blockDim.x + threadIdx.x) >> 5);
  const int n  = wave % NPTS;
  const int bf = wave / NPTS;
  const int m  = idx[bf * NPTS + n];
  const float4 v = *(const float4*)(valP + (size_t)(bf * NPTS + n) * COUT + lane * 4);
  float* p = pooled + (size_t)(bf * M + m) * COUT + lane * 4;
  atomicAdd(p + 0, v.x);
  atomicAdd(p + 1, v.y);
  atomicAdd(p + 2, v.z);
  atomicAdd(p + 3, v.w);
  if (lane == 0) atomicAdd(&counts[bf * M + m], 1.0f);
}

// ---------------- K6: divide + unfold + GroupNorm statistics ----------------
__global__ __launch_bounds__(256) void k_outpre(
    const float* __restrict__ pooled, const float* __restrict__ counts,
    float* __restrict__ outP, float* __restrict__ gnbuf) {
  const int i = (int)(blockIdx.x * blockDim.x + threadIdx.x); // [b][c][I][J]
  int t = i;
  const int J = t % OW; t /= OW;
  const int I = t % OH; t /= OH;
  const int c = t % COUT;
  const int b = t / COUT;
  const int f1 = I / CW, ci = I % CW;
  const int f2 = J / CW, cj = J % CW;
  const int bf = b * 4 + f1 * 2 + f2;
  const int m  = ci * CW + cj;
  const float val = pooled[(size_t)(bf * M + m) * COUT + c] / counts[bf * M + m];
  outP[i] = val;

  // block-wide sum/sumsq (each block lies entirely within one b)
  float s1 = waveReduceSum(val);
  float s2 = waveReduceSum(val * val);
  __shared__ float a1[8], a2[8];
  const int wid = threadIdx.x >> 5, lane = threadIdx.x & 31;
  if (lane == 0) { a1[wid] = s1; a2[wid] = s2; }
  __syncthreads();
  if (threadIdx.x == 0) {
    float t1 = 0.0f, t2 = 0.0f;
#pragma unroll
    for (int q = 0; q < 8; ++q) { t1 += a1[q]; t2 += a2[q]; }
    atomicAdd(&gnbuf[2 * b + 0], t1);
    atomicAdd(&gnbuf[2 * b + 1], t2);
  }
}

// ---------------- K7: 3x3 stride-2 skip conv + GroupNorm apply + add ----------------
__global__ __launch_bounds__(256) void k_final(
    const float* __restrict__ x, const float* __restrict__ wskip,
    const float* __restrict__ bskip, const float* __restrict__ gamma,
    const float* __restrict__ beta, const float* __restrict__ outP,
    const float* __restrict__ gnbuf, float* __restrict__ out) {
  const int bc    = blockIdx.x / 9;   // (b, c)
  const int chunk = blockIdx.x % 9;
  const int b = bc / COUT, c = bc % COUT;

  __shared__ float wsm[CIN * 9];
  for (int t = threadIdx.x; t < CIN * 9; t += 256)
    wsm[t] = wskip[(size_t)c * CIN * 9 + t];
  __syncthreads();

  const int pix = chunk * 256 + (int)threadIdx.x;  // 0..2303
  const int I = pix / OW, J = pix % OW;
  const int h0 = 2 * I - 1, w0 = 2 * J - 1;

  float acc = bskip[c];
  const float* xb = x + (size_t)b * CIN * H * W;
  for (int cin = 0; cin < CIN; ++cin) {
    const float* xc = xb + (size_t)cin * H * W;
    const float* wc = wsm + cin * 9;
#pragma unroll
    for (int di = 0; di < 3; ++di) {
      const int hh = h0 + di;
      if ((unsigned)hh >= (unsigned)H) continue;
#pragma unroll
      for (int dj = 0; dj < 3; ++dj) {
        const int ww = w0 + dj;
        if ((unsigned)ww >= (unsigned)W) continue;
        acc += xc[hh * W + ww] * wc[di * 3 + dj];
      }
    }
  }

  constexpr float invN = 1.0f / (float)(COUT * OH * OW);
  const float s1 = gnbuf[2 * b + 0];
  const float s2 = gnbuf[2 * b + 1];
  const float mu  = s1 * invN;
  const float var = fmaxf(s2 * invN - mu * mu, 0.0f);
  const size_t oidx = ((size_t)(b * COUT + c) * OH + I) * OW + J;
  const float g = (outP[oidx] - mu) * rsqrtf(var + 1e-5f);
  out[oidx] = acc + g * gamma[c] + beta[c];
}

// ---------------- launcher ----------------
extern "C" void kernel_launch(void* const* d_in, const int* in_sizes, int n_in,
                              void* d_out, int out_size, void* d_ws, size_t ws_size,
                              hipStream_t stream) {
  const float* x     = (const float*)d_in[0];
  const float* w_f   = (const float*)d_in[1];
  const float* b_f   = (const float*)d_in[2];
  const float* w_v   = (const float*)d_in[3];
  const float* b_v   = (const float*)d_in[4];
  const float* wskip = (const float*)d_in[5];
  const float* bskip = (const float*)d_in[6];
  const float* gamma = (const float*)d_in[7];
  const float* beta  = (const float*)d_in[8];

  float* ws     = (float*)d_ws;
  float* featP  = ws + OFF_FEAT;
  float* valP   = ws + OFF_VAL;
  float* cenN   = ws + OFF_CEN;
  float* vcen   = ws + OFF_VCEN;
  float* pooled = ws + OFF_POOL;
  float* counts = ws + OFF_CNT;
  float* gnbuf  = ws + OFF_GN;
  int*   idxp   = (int*)(ws + OFF_IDX);
  float* outP   = ws + OFF_FEAT;  // reuse featP region after k_sim

  const dim3 blk(256);

  // 1) feat/value = 1x1 convs (WMMA f32): BF*NT*OT waves, 8 waves/block
  k_gemm_fv<<<(BF * NT * OT) / 8, blk, 0, stream>>>(x, w_f, b_f, w_v, b_v, featP, valP);
  // 2) centers + value_centers (one wave per (bf, m))
  k_centers<<<(BF * M) / 8, blk, 0, stream>>>(featP, valP, cenN, vcen);
  // 3) sim GEMM + argmax (one wave per 16-point tile)
  k_sim<<<(BF * NT) / 8, blk, 0, stream>>>(featP, cenN, idxp);
  // 4) init pooled/counts/gn stats
  {
    const size_t n = SZ_CEN + (size_t)BF * M + 32;
    k_init<<<(unsigned)((n + 255) / 256), blk, 0, stream>>>(vcen, pooled, counts, gnbuf);
  }
  // 5) segment-sum scatter (one wave per (bf, n))
  k_scatter<<<(BF * NPTS) / 8, blk, 0, stream>>>(valP, idxp, pooled, counts);
  // 6) divide + unfold + GN stats (one thread per output element)
  k_outpre<<<(B0 * COUT * OH * OW) / 256, blk, 0, stream>>>(pooled, counts, outP, gnbuf);
  // 7) skip conv + GN apply + add
  k_final<<<B0 * COUT * 9, blk, 0, stream>>>(x, wskip, bskip, gamma, beta, outP, gnbuf,
                                             (float*)d_out);
}